// BasicEmbedding_62758062129575
// MI455X (gfx1250) — compile-verified
//
#include <hip/hip_runtime.h>
#include <hip/hip_bf16.h>
#include <cstdint>

// ---------------------------------------------------------------------------
// Embedding gather: out[t, :] = W[:, idx[t]] + b
//   idx: [262144] int32, W: [128, 100000] f32, b: [128] f32, out: [262144,128]
// Memory-bound (~190-290 MB @ 23.3 TB/s ~ 10us). Strategy:
//   1) transpose W -> WT[100000][128] in d_ws: the CDNA5 async global->LDS
//      copy does the transpose itself via the per-lane LDS destination
//      address (tracked by ASYNCcnt), then coalesced b128 stores.
//   2) wave-per-token gather of 512B rows from L2-resident WT (51MB << 192MB
//      L2), non-temporal b128 stores for the 134MB streaming output so the
//      write stream does not evict WT.
// ---------------------------------------------------------------------------

#define EMBED 128
#define NLAB  100000

typedef float v4f __attribute__((ext_vector_type(4)));

#if __has_builtin(__builtin_amdgcn_global_load_async_to_lds_b32) && \
    __has_builtin(__builtin_amdgcn_s_wait_asynccnt)
#define HAVE_ASYNC_LDS 1
typedef __attribute__((address_space(1))) int gas_int;
typedef __attribute__((address_space(3))) int las_int;
#endif

// ---------------------------------------------------------------------------
// Kernel 1: transpose W[128][100000] -> WT[100000][128].
// Tile: 128 rows x 32 cols per block. 100000 = 3125 * 32 exactly (no edges).
// LDS layout tile[col][e] with stride 129 -> conflict-free both phases:
//   write: addr = lane*129 + e  -> bank (lane + e) % 64, all distinct
//   read : 32 lanes read 128 consecutive floats of one row -> 2x each bank
// ---------------------------------------------------------------------------
__global__ __launch_bounds__(256) void
transpose_w_kernel(const float* __restrict__ W, float* __restrict__ WT) {
    __shared__ float tile[32 * 129];

    const int c0   = blockIdx.x * 32;        // first column of this tile
    const int t    = threadIdx.x;            // 0..255 (8 waves)
    const int lane = t & 31;                 // column within tile
    const int wv   = t >> 5;                 // 0..7

#ifdef HAVE_ASYNC_LDS
    // 16 straight-line async copies per thread, all in flight before the
    // single wait. Constant trip count -> full unroll, no loop masking.
    #pragma unroll
    for (int k = 0; k < EMBED / 8; ++k) {
        const int e = wv + 8 * k;
        __builtin_amdgcn_global_load_async_to_lds_b32(
            (gas_int*)(W + (size_t)e * NLAB + c0 + lane),
            (las_int*)(&tile[lane * 129 + e]),
            /*offset=*/0, /*cpol=*/0);
    }
    __builtin_amdgcn_s_wait_asynccnt(0);
    __syncthreads();
#else
    #pragma unroll
    for (int k = 0; k < EMBED / 8; ++k) {
        const int e = wv + 8 * k;
        tile[lane * 129 + e] = W[(size_t)e * NLAB + c0 + lane];
    }
    __syncthreads();
#endif

    // Write out: wave wv writes rows r = wv, wv+8, wv+16, wv+24 of the tile;
    // each row is 128 contiguous floats -> one b128 store per lane.
    #pragma unroll
    for (int r0 = 0; r0 < 4; ++r0) {
        const int r = wv + 8 * r0;
        v4f v;
        const float* src = &tile[r * 129 + lane * 4];
        v.x = src[0]; v.y = src[1]; v.z = src[2]; v.w = src[3];
        *(v4f*)(WT + (size_t)(c0 + r) * EMBED + lane * 4) = v;  // keep in L2
    }
}

// ---------------------------------------------------------------------------
// Kernel 2: gather rows of WT + bias, streamed out with NT stores.
// One wave per 32-token tile: coalesced idx load, bpermute-broadcast each
// index, whole wave reads one 512B row (b128/lane), writes one 512B out row.
// ---------------------------------------------------------------------------
__global__ __launch_bounds__(256) void
gather_rows_kernel(const int* __restrict__ idx, const float* __restrict__ WT,
                   const float* __restrict__ bias, float* __restrict__ out,
                   int ntok) {
    const int lane   = threadIdx.x & 31;
    const int wave   = blockIdx.x * (blockDim.x >> 5) + (threadIdx.x >> 5);
    const int nwaves = gridDim.x * (blockDim.x >> 5);

    const v4f bv = *(const v4f*)(bias + lane * 4);   // bias resident in VGPRs

    for (int t0 = wave * 32; t0 < ntok; t0 += nwaves * 32) {
        // Prefetch next idx tile (speculative; OOB is silently dropped).
        __builtin_prefetch(idx + t0 + (size_t)nwaves * 32 + lane, 0, 0);

        if (t0 + 32 <= ntok) {
            // Fast path: full tile, unrolled so row loads pipeline deeply.
            const int myidx = idx[t0 + lane];
            #pragma unroll 8
            for (int j = 0; j < 32; ++j) {
                const int row = __shfl(myidx, j, 32);
                v4f v = *(const v4f*)(WT + (size_t)row * EMBED + lane * 4);
                v += bv;
                __builtin_nontemporal_store(
                    v, (v4f*)(out + (size_t)(t0 + j) * EMBED + lane * 4));
            }
        } else {
            const int nt = ntok - t0;
            int myidx = 0;
            if (lane < nt) myidx = idx[t0 + lane];
            for (int j = 0; j < nt; ++j) {
                const int row = __shfl(myidx, j, 32);
                v4f v = *(const v4f*)(WT + (size_t)row * EMBED + lane * 4);
                v += bv;
                __builtin_nontemporal_store(
                    v, (v4f*)(out + (size_t)(t0 + j) * EMBED + lane * 4));
            }
        }
    }
}

// ---------------------------------------------------------------------------
// Fallback (ws too small): direct strided gather from W.
// ---------------------------------------------------------------------------
__global__ __launch_bounds__(256) void
gather_direct_kernel(const int* __restrict__ idx, const float* __restrict__ W,
                     const float* __restrict__ bias, float* __restrict__ out,
                     int ntok) {
    const int lane   = threadIdx.x & 31;
    const int wave   = blockIdx.x * (blockDim.x >> 5) + (threadIdx.x >> 5);
    const int nwaves = gridDim.x * (blockDim.x >> 5);

    for (int t = wave; t < ntok; t += nwaves) {
        const int c = idx[t];
        #pragma unroll
        for (int k = 0; k < 4; ++k) {
            const int e = lane + 32 * k;
            const float v = W[(size_t)e * NLAB + c] + bias[e];
            __builtin_nontemporal_store(v, out + (size_t)t * EMBED + e);
        }
    }
}

extern "C" void kernel_launch(void* const* d_in, const int* in_sizes, int n_in,
                              void* d_out, int out_size, void* d_ws, size_t ws_size,
                              hipStream_t stream) {
    const int*   X = (const int*)d_in[0];     // [32*8192] indices
    const float* W = (const float*)d_in[1];   // [128, 100000]
    const float* b = (const float*)d_in[2];   // [128]
    float*     out = (float*)d_out;           // [ntok, 128]
    const int ntok = in_sizes[0];

    const size_t wt_bytes = (size_t)NLAB * EMBED * sizeof(float);  // 51.2 MB

    if (ws_size >= wt_bytes) {
        float* WT = (float*)d_ws;
        transpose_w_kernel<<<NLAB / 32, 256, 0, stream>>>(W, WT);
        const int nblocks = (ntok + 256 - 1) / 256;   // 8 waves * 32 tok = 256/blk
        gather_rows_kernel<<<nblocks, 256, 0, stream>>>(X, WT, b, out, ntok);
    } else {
        const int nwaves  = ntok;                       // wave per token
        int nblocks = (nwaves + 7) / 8;
        if (nblocks > 4096) nblocks = 4096;
        gather_direct_kernel<<<nblocks, 256, 0, stream>>>(X, W, b, out, ntok);
    }
}